// DenseCapsule_17849884082709
// MI455X (gfx1250) — compile-verified
//
#include <hip/hip_runtime.h>

typedef __attribute__((ext_vector_type(2))) float v2f;
typedef __attribute__((ext_vector_type(8))) float v8f;

#define B_TOT 256
#define I_TOT 1152
#define K_TOT 8
#define J_TOT 10
#define D_TOT 16
#define EPSILON_F 1e-8f

// -----------------------------------------------------------------------------
// Kernel A: x_hat[b,j,i,d] = sum_k W[j,i,d,k] * x[b,i,k]
// One wave handles one (i, b_tile) and loops over all J=10 output capsules:
//   C_j[16b x 16d] += A[16b x 4k] x B_j[4k x 16d]   (two V_WMMA_F32_16X16X4_F32)
// The A fragments (x tile) are loaded ONCE and reused across all 10 j's.
// A layout (16x4 f32): lanes 0-15 -> M=0..15, k={0,1}; lanes 16-31 -> k={2,3}
// B layout (4x16 f32): lanes 0-15 -> N=0..15, k={0,1}; lanes 16-31 -> k={2,3}
// C layout (16x16 f32): vgpr r: lanes 0-15 -> M=r, lanes 16-31 -> M=r+8; N=lane%16
// -----------------------------------------------------------------------------
__global__ __launch_bounds__(256) void xhat_wmma_kernel(
    const float* __restrict__ x,      // (B, I, K)
    const float* __restrict__ w,      // (J, I, D, K)
    float* __restrict__ xhat)         // (B, J, I, D)
{
    const int lane = threadIdx.x & 31;
    const int wave = threadIdx.x >> 5;
    const int i    = blockIdx.x * 8 + wave;   // 1152/8 = 144 blocks in x
    const int b0   = blockIdx.y * 16;         // 256/16 = 16 blocks in y
    const int half = lane >> 4;
    const int lm   = lane & 15;

    // A fragments: load x once, reuse for all 10 j's.
    const float* xp = x + (size_t)(b0 + lm) * (I_TOT * K_TOT)
                        + (size_t)i * K_TOT + 2 * half;
    v2f a0, a1;
    a0.x = xp[0]; a0.y = xp[1];      // k = 2*half + {0,1}
    a1.x = xp[4]; a1.y = xp[5];      // k = 4 + 2*half + {0,1}

    const float* wp = w + ((size_t)i * D_TOT + lm) * K_TOT + 2 * half;
    float* op = xhat + (((size_t)(b0 + 8 * half) * J_TOT) * I_TOT + i) * D_TOT + lm;
    const size_t bstride = (size_t)J_TOT * I_TOT * D_TOT;   // stride in b
    const size_t jstrideW = (size_t)I_TOT * D_TOT * K_TOT;  // W stride in j
    const size_t jstrideO = (size_t)I_TOT * D_TOT;          // xhat stride in j

#pragma unroll
    for (int j = 0; j < J_TOT; ++j) {
        const float* wpj = wp + (size_t)j * jstrideW;
        v2f bm0, bm1;
        bm0.x = wpj[0]; bm0.y = wpj[1];
        bm1.x = wpj[4]; bm1.y = wpj[5];

        v8f c = {};
        c = __builtin_amdgcn_wmma_f32_16x16x4_f32(
                false, a0, false, bm0, (short)0, c, false, false);
        c = __builtin_amdgcn_wmma_f32_16x16x4_f32(
                false, a1, false, bm1, (short)0, c, false, false);

        float* opj = op + (size_t)j * jstrideO;
#pragma unroll
        for (int r = 0; r < 8; ++r) {
            opj[(size_t)r * bstride] = c[r];
        }
    }
}

// -----------------------------------------------------------------------------
// Kernel B: dynamic routing, fully fused per batch element b.
// logits/coeff live in LDS (I x J = 46 KB each); x_hat streamed from L2.
// -----------------------------------------------------------------------------
#define NCHUNK 3
#define CHUNK_I (I_TOT / NCHUNK)   // 384

__global__ __launch_bounds__(512) void routing_kernel(
    const float* __restrict__ xhat,   // (B, J, I, D)
    float* __restrict__ out)          // (B, J, D)
{
    __shared__ float logits[I_TOT][J_TOT];            // 46080 B
    __shared__ float coeff [I_TOT][J_TOT];            // 46080 B
    __shared__ float s_part[NCHUNK][J_TOT][D_TOT];    // 1920 B
    __shared__ float s_sh[J_TOT][D_TOT];
    __shared__ float v_sh[J_TOT][D_TOT];
    __shared__ float nrm[J_TOT];

    const int b   = blockIdx.x;
    const int tid = threadIdx.x;
    const float* xb = xhat + (size_t)b * J_TOT * I_TOT * D_TOT;

    for (int idx = tid; idx < I_TOT * J_TOT; idx += (int)blockDim.x) {
        logits[idx / J_TOT][idx % J_TOT] = 0.0f;
    }
    __syncthreads();

    for (int iter = 0; iter < 3; ++iter) {
        // coeff[i][:] = softmax over j of logits[i][:]
        for (int i = tid; i < I_TOT; i += (int)blockDim.x) {
            float m = logits[i][0];
#pragma unroll
            for (int j = 1; j < J_TOT; ++j) m = fmaxf(m, logits[i][j]);
            float e[J_TOT];
            float sum = 0.0f;
#pragma unroll
            for (int j = 0; j < J_TOT; ++j) {
                e[j] = __expf(logits[i][j] - m);
                sum += e[j];
            }
            const float inv = 1.0f / sum;
#pragma unroll
            for (int j = 0; j < J_TOT; ++j) coeff[i][j] = e[j] * inv;
        }
        __syncthreads();

        // s[j][d] = sum_i coeff[i][j] * xhat[b,j,i,d]
        // 480 threads: 3 chunks of I, 160 (j,d) lanes per chunk (coalesced over d)
        if (tid < NCHUNK * J_TOT * D_TOT) {
            const int g = tid / (J_TOT * D_TOT);
            const int r = tid % (J_TOT * D_TOT);
            const int j = r / D_TOT, d = r % D_TOT;
            const int i0 = g * CHUNK_I;
            const float* xp = xb + ((size_t)j * I_TOT + i0) * D_TOT + d;
            float acc = 0.0f;
            for (int i = 0; i < CHUNK_I; ++i)
                acc += coeff[i0 + i][j] * xp[(size_t)i * D_TOT];
            s_part[g][j][d] = acc;
        }
        __syncthreads();

        if (tid < J_TOT * D_TOT) {
            const int j = tid / D_TOT, d = tid % D_TOT;
            s_sh[j][d] = s_part[0][j][d] + s_part[1][j][d] + s_part[2][j][d];
        }
        __syncthreads();

        if (tid < J_TOT) {
            float n2 = 0.0f;
#pragma unroll
            for (int d = 0; d < D_TOT; ++d) n2 += s_sh[tid][d] * s_sh[tid][d];
            nrm[tid] = sqrtf(n2);
        }
        __syncthreads();

        // v = (s / (n + eps)) * (n^2 / (1 + n^2))
        if (tid < J_TOT * D_TOT) {
            const int j = tid / D_TOT, d = tid % D_TOT;
            const float n = nrm[j];
            const float scale = (n * n) / ((1.0f + n * n) * (n + EPSILON_F));
            v_sh[j][d] = s_sh[j][d] * scale;
        }
        __syncthreads();

        if (iter < 2) {
            // logits[i][j] += sum_d v[j][d] * xhat[b,j,i,d]  (float4 over d)
            for (int idx = tid; idx < I_TOT * J_TOT; idx += (int)blockDim.x) {
                const int i = idx / J_TOT, j = idx % J_TOT;
                const float4* xp = (const float4*)(xb + ((size_t)j * I_TOT + i) * D_TOT);
                const float4* vp = (const float4*)(&v_sh[j][0]);
                float acc = 0.0f;
#pragma unroll
                for (int q = 0; q < 4; ++q) {
                    const float4 xv = xp[q];
                    const float4 vv = vp[q];
                    acc += xv.x * vv.x + xv.y * vv.y + xv.z * vv.z + xv.w * vv.w;
                }
                logits[i][j] += acc;
            }
            __syncthreads();
        }
    }

    if (tid < J_TOT * D_TOT) {
        out[(size_t)b * (J_TOT * D_TOT) + tid] = v_sh[tid / D_TOT][tid % D_TOT];
    }
}

extern "C" void kernel_launch(void* const* d_in, const int* in_sizes, int n_in,
                              void* d_out, int out_size, void* d_ws, size_t ws_size,
                              hipStream_t stream) {
    const float* x = (const float*)d_in[0];   // (256, 1152, 8)
    const float* w = (const float*)d_in[1];   // (10, 1152, 16, 8)
    float* xhat = (float*)d_ws;               // (256, 10, 1152, 16) f32 = 189 MB

    dim3 gridA(I_TOT / 8, B_TOT / 16);        // 144 x 16 blocks, 8 waves each
    xhat_wmma_kernel<<<gridA, 256, 0, stream>>>(x, w, xhat);

    routing_kernel<<<B_TOT, 512, 0, stream>>>(xhat, (float*)d_out);
}